// DeepHopfield_43997644981023
// MI455X (gfx1250) — compile-verified
//
#include <hip/hip_runtime.h>
#include <math.h>

// ---------------------------------------------------------------------------
// DeepHopfield pipeline for MI455X (gfx1250), wave32.
// - FC GEMMs use V_WMMA_F32_16X16X4_F32 (memory-bound on f32 weights, so f32
//   WMMA is the roofline-correct precision; BW floor ~103MB @ 23.3 TB/s).
// - Enc-FC: split-K (196 chunks) for wave parallelism; A staged to LDS via
//   the Tensor Data Mover (tensor_load_to_lds + s_wait_tensorcnt). The LDS
//   pointer is laundered through inline asm so the compiler cannot fold the
//   reads of the DMA-written (source-invisible) LDS data.
// - Dec-FC: one wave per N-tile computes 4 M-tiles -> weights read once.
// - Hopfield recurrence: persistent single workgroup, WMMA + deterministic
//   shuffle-based energy reduction (no float atomics => bitwise stable).
// ---------------------------------------------------------------------------

typedef __attribute__((ext_vector_type(2))) float v2f;
typedef __attribute__((ext_vector_type(8))) float v8f;
typedef __attribute__((ext_vector_type(4))) unsigned int u32x4;
typedef __attribute__((ext_vector_type(8))) int i32x8;
typedef __attribute__((ext_vector_type(4))) int i32x4;

__device__ __forceinline__ float sgnf(float v) {
    return (float)((v > 0.f) - (v < 0.f));
}

// ---------------------------------------------------------------------------
// Direct 5x5 conv, pad=2, 28x28, one thread per output element.
// WMODE 0: weights [OC,IC,5,5] (normal conv)
// WMODE 1: weights [IC,OC,5,5], flipped taps (== torch ConvTranspose2d pad=2)
// ---------------------------------------------------------------------------
template <int IC, int OC, int WMODE, bool RELU>
__global__ void conv5x5_kernel(const float* __restrict__ in,
                               const float* __restrict__ w,
                               const float* __restrict__ bias,
                               float* __restrict__ out, int nTotal) {
    int idx = blockIdx.x * blockDim.x + threadIdx.x;
    if (idx >= nTotal) return;
    int x = idx % 28;
    int y = (idx / 28) % 28;
    int o = (idx / 784) % OC;
    int b = idx / (784 * OC);

    const float* ib = in + (size_t)b * IC * 784;
    float acc = bias[o];
    for (int i = 0; i < IC; ++i) {
        const float* ic = ib + i * 784;
        #pragma unroll
        for (int r = 0; r < 5; ++r) {
            int yy = y + r - 2;
            if (yy < 0 || yy >= 28) continue;
            #pragma unroll
            for (int s = 0; s < 5; ++s) {
                int xx = x + s - 2;
                if (xx < 0 || xx >= 28) continue;
                float wv;
                if (WMODE == 0)
                    wv = w[((o * IC + i) * 25) + r * 5 + s];
                else
                    wv = w[((i * OC + o) * 25) + (4 - r) * 5 + (4 - s)];
                acc = fmaf(ic[yy * 28 + xx], wv, acc);
            }
        }
    }
    if (RELU) acc = fmaxf(acc, 0.f);
    out[idx] = acc;
}

// ---------------------------------------------------------------------------
// Encoder FC, split-K:  pC[kc] = A[:, kc*256 : kc*256+256] @ B_chunk^T
// A: [64, 50176] (activations), B: [256, 50176] (enc_fc_w), pC: [196][64][256]
// One block per K-chunk. A chunk (64x256 f32 = 64KB) is DMA'd into LDS by
// the Tensor Data Mover; 8 waves each handle 2 N-tiles x 4 M-tiles.
// ---------------------------------------------------------------------------
__global__ void enc_fc_wmma_kernel(const float* __restrict__ A,
                                   const float* __restrict__ B,
                                   float* __restrict__ pC) {
    const int kc = blockIdx.x;  // 0..195
    const int wave = threadIdx.x >> 5;
    const int lane = threadIdx.x & 31;
    const int hh = lane >> 4;
    const int ll = lane & 15;

    __shared__ __align__(16) float As[64 * 256];

    if (wave == 0) {
        // Tensor DMA descriptor: 2D tile 256(x) x 64(y), 4-byte elements,
        // tensor row stride 50176, dense into LDS at offset 0.
        unsigned long long ga =
            (unsigned long long)(uintptr_t)(A + (size_t)kc * 256);
        u32x4 g0 = {1u,                                   // count=1
                    0u,                                   // lds_addr
                    (unsigned int)(ga & 0xFFFFFFFFu),
                    (unsigned int)(((ga >> 32) & 0x01FFFFFFu) | 0x80000000u)};
        i32x8 g1 = {(int)(2u << 16),                      // data_size=4B
                    (int)((50176u & 0xFFFFu) << 16),      // tensor_dim0 lo16
                    (int)(((50176u >> 16) & 0xFFFFu) | (64u << 16)),
                    (int)(256u << 16),                    // tile_dim0=256
                    64,                                   // tile_dim1=64
                    50176,                                // dim0_stride lo32
                    0, 0};
        i32x4 gz4 = {0, 0, 0, 0};
        i32x8 gz8 = {0, 0, 0, 0, 0, 0, 0, 0};
        __builtin_amdgcn_tensor_load_to_lds(g0, g1, gz4, gz4, gz8, 0);
        __builtin_amdgcn_s_wait_tensorcnt(0);
    }
    __syncthreads();

    // The TDM wrote As behind the compiler's back (the LDS address lives in
    // descriptor *data*). Escape the pointer through an asm with a memory
    // clobber so the A-fragment loads below cannot be folded away.
    const float* abase;
    {
        float* p = (float*)As;
        asm volatile("" : "+v"(p) : : "memory");
        abase = p;
    }

    #pragma unroll
    for (int sub = 0; sub < 2; ++sub) {
        const int nt = wave * 2 + sub;           // 0..15
        const int nBase = nt << 4;
        const float* brow =
            B + (size_t)(nBase + ll) * 50176 + (size_t)kc * 256 + 2 * hh;
        const float* alds = abase + ll * 256 + 2 * hh;  // + mt*16*256 + k

        v8f c0 = {0.f, 0.f, 0.f, 0.f, 0.f, 0.f, 0.f, 0.f};
        v8f c1 = c0, c2 = c0, c3 = c0;
        #pragma unroll 2
        for (int k = 0; k < 256; k += 4) {
            v2f b = *(const v2f*)(brow + k);
            v2f a0 = *(const v2f*)(alds + k);
            v2f a1 = *(const v2f*)(alds + 16 * 256 + k);
            v2f a2 = *(const v2f*)(alds + 32 * 256 + k);
            v2f a3 = *(const v2f*)(alds + 48 * 256 + k);
            c0 = __builtin_amdgcn_wmma_f32_16x16x4_f32(false, a0, false, b,
                                                       (short)0, c0, false, false);
            c1 = __builtin_amdgcn_wmma_f32_16x16x4_f32(false, a1, false, b,
                                                       (short)0, c1, false, false);
            c2 = __builtin_amdgcn_wmma_f32_16x16x4_f32(false, a2, false, b,
                                                       (short)0, c2, false, false);
            c3 = __builtin_amdgcn_wmma_f32_16x16x4_f32(false, a3, false, b,
                                                       (short)0, c3, false, false);
        }
        const int n = nBase + ll;
        float* pbase = pC + (size_t)kc * 64 * 256 + n;
        #pragma unroll
        for (int r = 0; r < 8; ++r) {
            int mr = r + 8 * hh;
            pbase[(size_t)(mr + 0) * 256]  = c0[r];
            pbase[(size_t)(mr + 16) * 256] = c1[r];
            pbase[(size_t)(mr + 32) * 256] = c2[r];
            pbase[(size_t)(mr + 48) * 256] = c3[r];
        }
    }
}

// z[m,n] = tanh( sum_kc pC[kc][m][n] + bias[n] )   (fixed order => deterministic)
__global__ void encfc_reduce_kernel(const float* __restrict__ pC,
                                    const float* __restrict__ bias,
                                    float* __restrict__ z) {
    int t = blockIdx.x * blockDim.x + threadIdx.x;
    if (t >= 64 * 256) return;
    int n = t & 255;
    float s = 0.f;
    for (int kc = 0; kc < 196; ++kc) s += pC[(size_t)kc * 16384 + t];
    z[t] = tanhf(s + bias[n]);
}

// ---------------------------------------------------------------------------
// Decoder FC: C[64,50176] = A[64,256] @ B[50176,256]^T + bias.
// One wave per N-tile; B fragment loaded once, 4 M-tile WMMAs per K-step
// => weight traffic hits the 51MB read-once floor. 3136 waves.
// ---------------------------------------------------------------------------
__global__ void dec_fc_wmma_kernel(const float* __restrict__ A,
                                   const float* __restrict__ B,
                                   const float* __restrict__ bias,
                                   float* __restrict__ C) {
    const int wave = threadIdx.x >> 5;
    const int lane = threadIdx.x & 31;
    const int hh = lane >> 4;
    const int ll = lane & 15;
    const int nt = blockIdx.x * 8 + wave;
    if (nt >= 3136) return;  // uniform per wave

    const int nBase = nt << 4;
    const float* brow = B + (size_t)(nBase + ll) * 256 + 2 * hh;
    const float* alds = A + ll * 256 + 2 * hh;

    v8f c0 = {0.f, 0.f, 0.f, 0.f, 0.f, 0.f, 0.f, 0.f};
    v8f c1 = c0, c2 = c0, c3 = c0;
    #pragma unroll 2
    for (int k = 0; k < 256; k += 4) {
        __builtin_prefetch(brow + 256 + k, 0, 0);  // next B row, speculative
        v2f b = *(const v2f*)(brow + k);
        v2f a0 = *(const v2f*)(alds + k);
        v2f a1 = *(const v2f*)(alds + 16 * 256 + k);
        v2f a2 = *(const v2f*)(alds + 32 * 256 + k);
        v2f a3 = *(const v2f*)(alds + 48 * 256 + k);
        c0 = __builtin_amdgcn_wmma_f32_16x16x4_f32(false, a0, false, b, (short)0,
                                                   c0, false, false);
        c1 = __builtin_amdgcn_wmma_f32_16x16x4_f32(false, a1, false, b, (short)0,
                                                   c1, false, false);
        c2 = __builtin_amdgcn_wmma_f32_16x16x4_f32(false, a2, false, b, (short)0,
                                                   c2, false, false);
        c3 = __builtin_amdgcn_wmma_f32_16x16x4_f32(false, a3, false, b, (short)0,
                                                   c3, false, false);
    }
    const int n = nBase + ll;
    const float bv = bias[n];
    #pragma unroll
    for (int r = 0; r < 8; ++r) {
        int mr = r + 8 * hh;
        C[(size_t)(mr + 0) * 50176 + n]  = c0[r] + bv;
        C[(size_t)(mr + 16) * 50176 + n] = c1[r] + bv;
        C[(size_t)(mr + 32) * 50176 + n] = c2[r] + bv;
        C[(size_t)(mr + 48) * 50176 + n] = c3[r] + bv;
    }
}

// ---------------------------------------------------------------------------
// rho = mean(sign(label_latent)) over all 10*256 elements.
// ---------------------------------------------------------------------------
__global__ void rho_kernel(const float* __restrict__ ll, float* __restrict__ rho) {
    __shared__ float red[256];
    float s = 0.f;
    for (int i = threadIdx.x; i < 2560; i += 256) s += sgnf(ll[i]);
    red[threadIdx.x] = s;
    __syncthreads();
    for (int off = 128; off > 0; off >>= 1) {
        if (threadIdx.x < off) red[threadIdx.x] += red[threadIdx.x + off];
        __syncthreads();
    }
    if (threadIdx.x == 0) *rho = red[0] / 2560.f;
}

// w[i][j] = (i==j) ? 0 : sum_l (sgn(ll[l,i])-rho)*(sgn(ll[l,j])-rho) / 10
__global__ void hopw_kernel(const float* __restrict__ ll,
                            const float* __restrict__ rho,
                            float* __restrict__ w) {
    int i = blockIdx.x;
    int j = threadIdx.x;
    float r = *rho;
    float acc = 0.f;
    #pragma unroll
    for (int l = 0; l < 10; ++l) {
        float ti = sgnf(ll[l * 256 + i]) - r;
        float tj = sgnf(ll[l * 256 + j]) - r;
        acc = fmaf(ti, tj, acc);
    }
    w[i * 256 + j] = (i == j) ? 0.f : acc * 0.1f;
}

// ---------------------------------------------------------------------------
// Hopfield recurrence (single persistent workgroup of 8 waves).
// ---------------------------------------------------------------------------
__device__ __forceinline__ v8f tile_mm_256(const float* __restrict__ S,
                                           const float* __restrict__ W,
                                           int mBase, int nBase, int hh, int ll) {
    const float* arow = S + (mBase + ll) * 256 + 2 * hh;
    const float* brow = W + (nBase + ll) * 256 + 2 * hh;
    v8f c = {0.f, 0.f, 0.f, 0.f, 0.f, 0.f, 0.f, 0.f};
    #pragma unroll 8
    for (int k = 0; k < 256; k += 4) {
        v2f a = *(const v2f*)(arow + k);
        v2f b = *(const v2f*)(brow + k);
        c = __builtin_amdgcn_wmma_f32_16x16x4_f32(false, a, false, b, (short)0, c,
                                                  false, false);
    }
    return c;
}

// Deterministic per-row energy partials: ep[nt][m] = sum over the tile's 16
// columns of S[m,n]*(S@w)[m,n], reduced via shuffles within each 16-lane half.
__device__ __forceinline__ void energy_tiles(const float* __restrict__ S,
                                             const float* __restrict__ W,
                                             float* __restrict__ ep,  // [16][64]
                                             int wave, int hh, int ll) {
    for (int t = wave; t < 64; t += 8) {
        int mBase = (t >> 4) << 4, nt = t & 15, nBase = nt << 4;
        v8f c = tile_mm_256(S, W, mBase, nBase, hh, ll);
        float pr[8];
        #pragma unroll
        for (int r = 0; r < 8; ++r) {
            int m = mBase + r + 8 * hh, n = nBase + ll;
            pr[r] = S[m * 256 + n] * c[r];
        }
        #pragma unroll
        for (int mask = 1; mask < 16; mask <<= 1) {
            #pragma unroll
            for (int r = 0; r < 8; ++r) pr[r] += __shfl_xor(pr[r], mask, 32);
        }
        if (ll == 0) {  // lanes 0 and 16 hold their half's sums
            #pragma unroll
            for (int r = 0; r < 8; ++r)
                ep[nt * 64 + mBase + r + 8 * hh] = pr[r];
        }
    }
}

__global__ void cluster_kernel(const float* __restrict__ z,
                               const float* __restrict__ w,
                               float* __restrict__ Sa,
                               float* __restrict__ Sb) {
    __shared__ float ep[16 * 64];  // per-(nTile,row) energy partials
    __shared__ float e_prev[64];
    __shared__ int done[64];

    const int tid = threadIdx.x;
    const int wave = tid >> 5;
    const int lane = tid & 31;
    const int hh = lane >> 4;
    const int ll = lane & 15;

    // s0 = sign(z)
    for (int i = tid; i < 64 * 256; i += 256) Sa[i] = sgnf(z[i]);
    if (tid < 64) done[tid] = 0;
    __threadfence();
    __syncthreads();

    // initial energy of s0
    energy_tiles(Sa, w, ep, wave, hh, ll);
    __syncthreads();
    if (tid < 64) {
        float e = 0.f;
        #pragma unroll
        for (int nt = 0; nt < 16; ++nt) e += ep[nt * 64 + tid];
        e_prev[tid] = -e;
    }
    __syncthreads();

    for (int it = 0; it < 256; ++it) {
        const float* cur = (it & 1) ? Sb : Sa;
        float* nxt = (it & 1) ? Sa : Sb;

        // phase 1: nxt = done[m] ? cur : sign(cur @ w)   (w symmetric)
        for (int t = wave; t < 64; t += 8) {
            int mBase = (t >> 4) << 4, nBase = (t & 15) << 4;
            v8f c = tile_mm_256(cur, w, mBase, nBase, hh, ll);
            #pragma unroll
            for (int r = 0; r < 8; ++r) {
                int m = mBase + r + 8 * hh, n = nBase + ll;
                nxt[m * 256 + n] = done[m] ? cur[m * 256 + n] : sgnf(c[r]);
            }
        }
        __threadfence();
        __syncthreads();

        // phase 2: deterministic energy of nxt
        energy_tiles(nxt, w, ep, wave, hh, ll);
        __syncthreads();

        // phase 3: reference masking semantics (old done / old prev_e)
        if (tid < 64) {
            float e = 0.f;
            #pragma unroll
            for (int nt = 0; nt < 16; ++nt) e += ep[nt * 64 + tid];
            e = -e;
            if (!done[tid]) {
                if (e == e_prev[tid]) done[tid] = 1;
                e_prev[tid] = e;
            }
        }
        __syncthreads();
    }
    // 256 iterations (even) -> final state lives in Sa.
}

// ---------------------------------------------------------------------------
// Both softmax heads; one thread per batch row.
// ---------------------------------------------------------------------------
__global__ void probs_kernel(const float* __restrict__ S,
                             const float* __restrict__ ll,
                             const float* __restrict__ z,
                             const float* __restrict__ smw,
                             const float* __restrict__ smb,
                             float* __restrict__ out) {
    int b = blockIdx.x * blockDim.x + threadIdx.x;
    if (b >= 64) return;

    float lg[10];
    float mx = -1e30f;
    #pragma unroll
    for (int c = 0; c < 10; ++c) {
        float a = 0.f;
        for (int i = 0; i < 256; ++i) a = fmaf(S[b * 256 + i], ll[c * 256 + i], a);
        lg[c] = a;
        mx = fmaxf(mx, a);
    }
    float den = 0.f;
    #pragma unroll
    for (int c = 0; c < 10; ++c) { lg[c] = expf(lg[c] - mx); den += lg[c]; }
    #pragma unroll
    for (int c = 0; c < 10; ++c) out[b * 10 + c] = lg[c] / den;

    mx = -1e30f;
    #pragma unroll
    for (int c = 0; c < 10; ++c) {
        float a = smb[c];
        for (int i = 0; i < 256; ++i) a = fmaf(z[b * 256 + i], smw[c * 256 + i], a);
        lg[c] = a;
        mx = fmaxf(mx, a);
    }
    den = 0.f;
    #pragma unroll
    for (int c = 0; c < 10; ++c) { lg[c] = expf(lg[c] - mx); den += lg[c]; }
    #pragma unroll
    for (int c = 0; c < 10; ++c) out[640 + b * 10 + c] = lg[c] / den;
}

__global__ void atanh_kernel(const float* __restrict__ z, float* __restrict__ az,
                             int n) {
    int i = blockIdx.x * blockDim.x + threadIdx.x;
    if (i < n) az[i] = atanhf(z[i]);
}

// ---------------------------------------------------------------------------
// Host launcher
// ---------------------------------------------------------------------------
extern "C" void kernel_launch(void* const* d_in, const int* in_sizes, int n_in,
                              void* d_out, int out_size, void* d_ws,
                              size_t ws_size, hipStream_t stream) {
    (void)in_sizes; (void)n_in; (void)out_size; (void)ws_size;

    const float* images    = (const float*)d_in[0];
    const float* enc_c1_w  = (const float*)d_in[1];
    const float* enc_c1_b  = (const float*)d_in[2];
    const float* enc_c2_w  = (const float*)d_in[3];
    const float* enc_c2_b  = (const float*)d_in[4];
    const float* enc_fc_w  = (const float*)d_in[5];
    const float* enc_fc_b  = (const float*)d_in[6];
    const float* dec_fc_w  = (const float*)d_in[7];
    const float* dec_fc_b  = (const float*)d_in[8];
    const float* dec_ct2_w = (const float*)d_in[9];
    const float* dec_ct2_b = (const float*)d_in[10];
    const float* dec_ct1_w = (const float*)d_in[11];
    const float* dec_ct1_b = (const float*)d_in[12];
    const float* sm_w      = (const float*)d_in[13];
    const float* sm_b      = (const float*)d_in[14];
    const float* lab_lat   = (const float*)d_in[15];

    float* out = (float*)d_out;  // [640 cluster | 640 class | 50176 recon]
    float* ws  = (float*)d_ws;

    size_t off = 0;
    auto carve = [&](size_t n) { float* p = ws + off; off += n; return p; };
    float* buf1 = carve((size_t)64 * 32 * 784);
    float* buf2 = carve((size_t)64 * 64 * 784);
    float* pC   = carve((size_t)196 * 64 * 256);  // enc-FC split-K partials
    float* z    = carve(64 * 256);
    float* rho  = carve(64);
    float* hopw = carve(256 * 256);
    float* sa   = carve(64 * 256);
    float* sb   = carve(64 * 256);
    float* az   = carve(64 * 256);
    float* dfc  = carve((size_t)64 * 50176);
    float* ct2o = carve((size_t)64 * 32 * 784);
    // total scratch ~52 MB

    // --- encoder ---
    {
        int n = 64 * 32 * 784;
        conv5x5_kernel<1, 32, 0, true>
            <<<(n + 255) / 256, 256, 0, stream>>>(images, enc_c1_w, enc_c1_b, buf1, n);
    }
    {
        int n = 64 * 64 * 784;
        conv5x5_kernel<32, 64, 0, true>
            <<<(n + 255) / 256, 256, 0, stream>>>(buf1, enc_c2_w, enc_c2_b, buf2, n);
    }
    enc_fc_wmma_kernel<<<196, 256, 0, stream>>>(buf2, enc_fc_w, pC);
    encfc_reduce_kernel<<<64, 256, 0, stream>>>(pC, enc_fc_b, z);

    // --- Hopfield weight + clustering ---
    rho_kernel<<<1, 256, 0, stream>>>(lab_lat, rho);
    hopw_kernel<<<256, 256, 0, stream>>>(lab_lat, rho, hopw);
    cluster_kernel<<<1, 256, 0, stream>>>(z, hopw, sa, sb);

    // --- softmax heads ---
    probs_kernel<<<1, 64, 0, stream>>>(sa, lab_lat, z, sm_w, sm_b, out);

    // --- decoder ---
    atanh_kernel<<<64, 256, 0, stream>>>(z, az, 64 * 256);
    dec_fc_wmma_kernel<<<392, 256, 0, stream>>>(az, dec_fc_w, dec_fc_b, dfc);
    {
        int n = 64 * 32 * 784;
        conv5x5_kernel<64, 32, 1, true>
            <<<(n + 255) / 256, 256, 0, stream>>>(dfc, dec_ct2_w, dec_ct2_b, ct2o, n);
    }
    {
        int n = 64 * 1 * 784;
        conv5x5_kernel<32, 1, 1, false>
            <<<(n + 255) / 256, 256, 0, stream>>>(ct2o, dec_ct1_w, dec_ct1_b,
                                                  out + 1280, n);
    }
}